// LSTM_51840255262847
// MI455X (gfx1250) — compile-verified
//
#include <hip/hip_runtime.h>

// ---------------------------------------------------------------------------
// Fused 2-layer LSTM + output projection for gfx1250 (MI455X), fp32 WMMA.
//
// Persistent kernel, software-pipelined over timesteps (one grid barrier per
// iteration). 40 blocks x 256 threads (8 wave32):
//   blocks  0..15 : layer 0 (block j owns 16 H-cols, all gates; wave r owns
//                   16 batch rows). Weights LDS-resident; x_t double-buffered
//                   in LDS via GLOBAL_LOAD_ASYNC_TO_LDS_B128 overlapping the
//                   WMMA chains (ASYNCcnt pipelining).
//   blocks 16..31 : layer 1, same decomposition; weights LDS-resident.
//   blocks 32..39 : output projection (w_lin tile LDS-resident).
// c-state lives in VGPRs for the whole sequence; h double-buffers live in
// global workspace in fragment-major layout F[k>>2][row][k&3] so every WMMA
// A/B fragment fetch is one contiguous 256B span per wave. All streams use
// explicitly address-space-qualified pointers (global vs LDS) to guarantee
// global_load_b64 / ds_load_b64 (never flat_load).
// ---------------------------------------------------------------------------

typedef __attribute__((ext_vector_type(2))) float v2f;
typedef __attribute__((ext_vector_type(8))) float v8f;

typedef __attribute__((address_space(1))) float gfloat;
typedef __attribute__((address_space(3))) float lfloat;
typedef const gfloat* gcf;
typedef gfloat* gf;
typedef const lfloat* lcf;
typedef lfloat* lf;
typedef __attribute__((address_space(1))) v2f gv2f;
typedef __attribute__((address_space(3))) v2f lv2f;

#define NBLK 40
#define B_ 128
#define T_ 1024
#define IN_ 128
#define H_ 256
#define BH (B_ * H_)

__device__ __attribute__((always_inline)) inline v2f ld2(gcf p) {
  return *(const gv2f*)p;
}
__device__ __attribute__((always_inline)) inline v2f ld2(lcf p) {
  return *(const lv2f*)p;
}

__device__ __attribute__((always_inline)) inline v8f
wmma_f32(v2f a, v2f b, v8f c) {
  return __builtin_amdgcn_wmma_f32_16x16x4_f32(false, a, false, b,
                                               (short)0, c, false, false);
}

__device__ __attribute__((always_inline)) inline float sigmoid_f(float x) {
  return 1.0f / (1.0f + __expf(-x));
}

// Stage a weight tile (4 gates x 16 cols, K deep) into LDS, fragment-major
// with Rdim=64 (row = g*16 + col%16). Coalesced on k. One-time cost.
__device__ void stage_w(const float* __restrict__ w, lf wl, int cbase, int K) {
  for (int idx = threadIdx.x; idx < 64 * K; idx += 256) {
    int row = idx / K;  // g*16 + cl
    int k = idx - row * K;
    int g = row >> 4, cl = row & 15;
    wl[((k >> 2) * 64 + row) * 4 + (k & 3)] =
        w[(size_t)(g * H_ + cbase + cl) * K + k];
  }
}

// Issue async staging of x_t (128 rows x 128 feats) into LDS fragment-major
// as 16B packets (4 consecutive k contiguous in LDS and global). No wait:
// the consumer waits ASYNCcnt at the top of the next iteration.
__device__ __attribute__((always_inline)) void stage_x_issue(
    const float* __restrict__ xt, lf xf) {
#pragma unroll 4
  for (int idx = threadIdx.x; idx < (B_ * IN_) / 4; idx += 256) {
    int b = idx >> 5, kc = idx & 31;  // consecutive lanes -> consecutive kc
    const float* src = xt + (size_t)b * (T_ * IN_) + kc * 4;
    unsigned dst = (unsigned)(size_t)(xf + (kc * B_ + b) * 4);  // LDS offset
    asm volatile("global_load_async_to_lds_b128 %0, %1, off"
                 :: "v"(dst), "v"(src) : "memory");
    __builtin_prefetch(src + IN_, 0, 1);  // warm L2 for step t+1
  }
}

// One LSTM cell step for a 16(B) x 16(H) tile; all 4 gates.
//   xp:   input-activation frags (LDS for L0, global h0 buffer for L1),
//         chunk stride 512 floats (Rdim=128)
//   hp:   recurrent h frags (global), chunk stride 512
//   wp_*: LDS weight frags, gate stride 64, chunk stride 256 (Rdim=64)
//   hw:   per-lane write base into h_next frag buffer, element stride 4
template <int XCH, typename XPT>
__device__ __attribute__((always_inline)) void lstm_cell(
    XPT xp, gcf hp, lcf wp_ih, lcf wp_hh,
    const float* bsum, gf hw, v8f& ctile, v8f& htile) {
  v8f acc[4];
#pragma unroll
  for (int g = 0; g < 4; ++g) {
    float bs = bsum[g];
    v8f a = {bs, bs, bs, bs, bs, bs, bs, bs};
    acc[g] = a;
  }
  // input term: x_t @ w_ih^T (A fragment shared by 4 independent gate chains)
#pragma unroll 8
  for (int i = 0; i < XCH; ++i) {
    v2f a = ld2(xp + i * 512);
#pragma unroll
    for (int g = 0; g < 4; ++g)
      acc[g] = wmma_f32(a, ld2(wp_ih + g * 64 + i * 256), acc[g]);
  }
  // recurrent term: h_{t-1} @ w_hh^T
#pragma unroll 8
  for (int i = 0; i < 64; ++i) {
    v2f a = ld2(hp + i * 512);
#pragma unroll
    for (int g = 0; g < 4; ++g)
      acc[g] = wmma_f32(a, ld2(wp_hh + g * 64 + i * 256), acc[g]);
  }
  // pointwise LSTM update (gate order i,f,g,o); write h in fragment layout
#pragma unroll
  for (int e = 0; e < 8; ++e) {
    float ig = sigmoid_f(acc[0][e]);
    float fg = sigmoid_f(acc[1][e]);
    float gg = tanhf(acc[2][e]);
    float og = sigmoid_f(acc[3][e]);
    float c = fg * ctile[e] + ig * gg;
    ctile[e] = c;
    float h = og * tanhf(c);
    htile[e] = h;
    hw[e * 4] = h;
  }
}

__device__ __attribute__((always_inline)) void out_proj(
    gcf ap, lcf wlp, float bs, gf op) {
  // two independent accumulation chains to break the serial WMMA dependency
  v8f acc0 = {bs, bs, bs, bs, bs, bs, bs, bs};
  v8f acc1 = {0.f, 0.f, 0.f, 0.f, 0.f, 0.f, 0.f, 0.f};
#pragma unroll 8
  for (int i = 0; i < 64; i += 2) {
    acc0 = wmma_f32(ld2(ap + i * 512), ld2(wlp + i * 64), acc0);
    acc1 = wmma_f32(ld2(ap + (i + 1) * 512), ld2(wlp + (i + 1) * 64), acc1);
  }
#pragma unroll
  for (int e = 0; e < 8; ++e) op[(size_t)e * (T_ * IN_)] = acc0[e] + acc1[e];
}

__global__ __launch_bounds__(256, 1) void lstm_fused(
    const float* __restrict__ x,
    const float* __restrict__ w_ih0, const float* __restrict__ w_hh0,
    const float* __restrict__ b_ih0, const float* __restrict__ b_hh0,
    const float* __restrict__ w_ih1, const float* __restrict__ w_hh1,
    const float* __restrict__ b_ih1, const float* __restrict__ b_hh1,
    const float* __restrict__ w_lin, const float* __restrict__ b_lin,
    float* __restrict__ out, float* __restrict__ ws) {
  extern __shared__ float smem[];
  lf sm = (lf)smem;

  unsigned* bar_cnt = (unsigned*)ws;
  unsigned* bar_gen = bar_cnt + 1;
  gf h0buf = (gf)(ws + 64);       // [2][fragment-major B_ x H_]
  gf h1buf = h0buf + 2 * BH;
  gf outg = (gf)out;
  gf h_out = outg + (size_t)B_ * T_ * IN_;  // [2][B][H]
  gf c_out = h_out + 2 * BH;                // [2][B][H]

  const int lane = threadIdx.x & 31;
  const int wave = threadIdx.x >> 5;
  const int blk = blockIdx.x;
  const int rbase = wave * 16;  // batch-row tile for this wave
  const int l15 = lane & 15, lh = lane >> 4;

  // ---- LDS partition -------------------------------------------------------
  lf xf0 = sm;               // L0: x frag double buffer [32][128][4] x2 (128KB)
  lf xf1 = sm + 16384;
  lf wih_l = sm + 32768;     // L0: [32][64][4] (32 KB)
  lf whh_l = sm + 40960;     // L0: [64][64][4] (64 KB)

  // ---- role setup + one-time LDS weight staging ----------------------------
  float bsum[4] = {0.f, 0.f, 0.f, 0.f};
  float blin_s = 0.f;
  int cbase = 0;

  if (blk < 16) {
    cbase = blk * 16;
    stage_w(w_ih0, wih_l, cbase, IN_);
    stage_w(w_hh0, whh_l, cbase, H_);
    int nn = cbase + l15;
#pragma unroll
    for (int g = 0; g < 4; ++g)
      bsum[g] = b_ih0[g * H_ + nn] + b_hh0[g * H_ + nn];
  } else if (blk < 32) {
    cbase = (blk - 16) * 16;
    wih_l = sm;              // [64][64][4] (64 KB)
    whh_l = sm + 16384;      // [64][64][4] (64 KB)
    stage_w(w_ih1, wih_l, cbase, H_);
    stage_w(w_hh1, whh_l, cbase, H_);
    int nn = cbase + l15;
#pragma unroll
    for (int g = 0; g < 4; ++g)
      bsum[g] = b_ih1[g * H_ + nn] + b_hh1[g * H_ + nn];
  } else {
    cbase = (blk - 32) * 16;
    for (int idx = threadIdx.x; idx < 16 * H_; idx += 256) {
      int cl = idx >> 8, k = idx & 255;
      sm[((k >> 2) * 16 + cl) * 4 + (k & 3)] =
          w_lin[(size_t)(cbase + cl) * H_ + k];
    }
    blin_s = b_lin[cbase + l15];
  }
  __syncthreads();

  // ---- per-lane fragment base pointers (loop-invariant) --------------------
  const int rdoff = (rbase + l15) * 4 + 2 * lh;  // frag read offset, Rdim=128
  const int n = cbase + l15;
  const int wroff = ((n >> 2) * B_ + rbase + 8 * lh) * 4 + (n & 3);
  gcf h0r[2] = {h0buf + rdoff, h0buf + BH + rdoff};
  gcf h1r[2] = {h1buf + rdoff, h1buf + BH + rdoff};
  gf h0w[2] = {h0buf + wroff, h0buf + BH + wroff};
  gf h1w[2] = {h1buf + wroff, h1buf + BH + wroff};
  lcf xp[2] = {xf0 + rdoff, xf1 + rdoff};
  lf xfb[2] = {xf0, xf1};
  lcf wp_ih = wih_l + l15 * 4 + 2 * lh;
  lcf wp_hh = whh_l + l15 * 4 + 2 * lh;
  lcf wp_lin = sm + l15 * 4 + 2 * lh;

  v8f ct = {0.f, 0.f, 0.f, 0.f, 0.f, 0.f, 0.f, 0.f};
  v8f ht = {0.f, 0.f, 0.f, 0.f, 0.f, 0.f, 0.f, 0.f};

  // prologue: async-stage x_0 into buffer 0
  if (blk < 16) stage_x_issue(x, xfb[0]);

  // ---- time loop: L0 step it | L1 step it-1 | out-proj step it-2 ----------
  for (int it = 0; it < T_ + 2; ++it) {
    const int cur = it & 1;
    const int nxt = (it + 1) & 1;

    if (blk < 16) {
      if (it < T_) {
        // complete my async packets for step it, sync so all waves' packets
        // are visible, then issue step it+1 into the other buffer (overlaps
        // with this step's WMMA work).
        asm volatile("s_wait_asynccnt 0x0" ::: "memory");
        __syncthreads();
        if (it + 1 < T_)
          stage_x_issue(x + (size_t)(it + 1) * IN_, xfb[nxt]);
        lstm_cell<32>(xp[cur], h0r[cur], wp_ih, wp_hh, bsum, h0w[nxt], ct, ht);
      }
    } else if (blk < 32) {
      if ((unsigned)(it - 1) < (unsigned)T_)
        lstm_cell<64>(h0r[cur], h1r[cur], wp_ih, wp_hh, bsum, h1w[nxt], ct, ht);
    } else {
      int t = it - 2;
      if (t >= 0)
        out_proj(h1r[cur], wp_lin, blin_s,
                 outg + ((size_t)(rbase + 8 * lh) * T_ + t) * IN_ + n);
    }

    // ---- grid-wide barrier (self-resetting; gen == completed barriers) ----
    __threadfence();
    __syncthreads();
    if (threadIdx.x == 0) {
      unsigned old = __hip_atomic_fetch_add(bar_cnt, 1u, __ATOMIC_ACQ_REL,
                                            __HIP_MEMORY_SCOPE_AGENT);
      if (old == (unsigned)(NBLK - 1)) {
        __hip_atomic_store(bar_cnt, 0u, __ATOMIC_RELAXED,
                           __HIP_MEMORY_SCOPE_AGENT);
        __hip_atomic_store(bar_gen, (unsigned)(it + 1), __ATOMIC_RELEASE,
                           __HIP_MEMORY_SCOPE_AGENT);
      } else {
        while (__hip_atomic_load(bar_gen, __ATOMIC_ACQUIRE,
                                 __HIP_MEMORY_SCOPE_AGENT) <
               (unsigned)(it + 1)) {
          __builtin_amdgcn_s_sleep(2);
        }
      }
    }
    __syncthreads();
    __builtin_amdgcn_fence(__ATOMIC_ACQUIRE, "agent");
  }

  // ---- final (h_T, c_T) written once, from registers -----------------------
  if (blk < 32) {
    size_t off = (blk < 16) ? 0 : (size_t)BH;
#pragma unroll
    for (int e = 0; e < 8; ++e) {
      int m = e + 8 * lh;
      h_out[off + (size_t)(rbase + m) * H_ + n] = ht[e];
      c_out[off + (size_t)(rbase + m) * H_ + n] = ct[e];
    }
  }
}

extern "C" void kernel_launch(void* const* d_in, const int* in_sizes, int n_in,
                              void* d_out, int out_size, void* d_ws,
                              size_t ws_size, hipStream_t stream) {
  const float* x    = (const float*)d_in[0];
  const float* wih0 = (const float*)d_in[1];
  const float* whh0 = (const float*)d_in[2];
  const float* bih0 = (const float*)d_in[3];
  const float* bhh0 = (const float*)d_in[4];
  const float* wih1 = (const float*)d_in[5];
  const float* whh1 = (const float*)d_in[6];
  const float* bih1 = (const float*)d_in[7];
  const float* bhh1 = (const float*)d_in[8];
  const float* wlin = (const float*)d_in[9];
  const float* blin = (const float*)d_in[10];
  (void)in_sizes; (void)n_in; (void)out_size; (void)ws_size;

  const size_t smem_bytes = 57344 * sizeof(float);  // 224 KB (L0 blocks)
  hipFuncSetAttribute((const void*)lstm_fused,
                      hipFuncAttributeMaxDynamicSharedMemorySize,
                      (int)smem_bytes);

  // Zero barrier state + fragment-major h double-buffers (capture-safe).
  size_t ws_used = 64 * sizeof(float) + (size_t)4 * BH * sizeof(float);
  hipMemsetAsync(d_ws, 0, ws_used, stream);

  lstm_fused<<<NBLK, 256, smem_bytes, stream>>>(
      x, wih0, whh0, bih0, bhh0, wih1, whh1, bih1, bhh1, wlin, blin,
      (float*)d_out, (float*)d_ws);
}